// BiLSTM_CRF_9577777070207
// MI455X (gfx1250) — compile-verified
//
#include <hip/hip_runtime.h>
#include <math.h>

// BiLSTM-CRF forward. V=128, T=32, E=256, H=512, F=64, L=2, B=32, S=512, NH=4, D=1024, dh=256.
// fp32 everywhere (reference precision) -> V_WMMA_F32_16X16X4_F32 is the matrix path.

typedef float v2f __attribute__((ext_vector_type(2)));
typedef float v4f __attribute__((ext_vector_type(4)));
typedef float v8f __attribute__((ext_vector_type(8)));

// WMMA fp32 16x16x4 layouts (ISA 7.12.2, wave32):
//  A (16x4): lane L, reg j: M = L&15, K = (L>>4)*2 + j
//  B (4x16): lane L, reg j: N = L&15, K = (L>>4)*2 + j
//  C (16x16): lane L, reg r: N = L&15, M = r + (L>>4)*8
static __device__ __forceinline__ v8f wmma4(v2f a, v2f b, v8f c) {
  return __builtin_amdgcn_wmma_f32_16x16x4_f32(false, a, false, b, (short)0, c, false, false);
}
static __device__ __forceinline__ float sig_(float x) { return 1.0f / (1.0f + expf(-x)); }

// ---------------- workspace layout (float offsets) ----------------
static constexpr size_t OFF_X   = 0;                      // (B,S,512)
static constexpr size_t OFF_HB  = OFF_X  + 8388608;       // (S,B,256)
static constexpr size_t OFF_Y1  = OFF_HB + 4194304;       // (S,B,1024)
static constexpr size_t OFF_Y2  = OFF_Y1 + 16777216;      // (S,B,1024)
static constexpr size_t OFF_Q   = OFF_Y2 + 16777216;      // (B*NH,S,256)
static constexpr size_t OFF_K   = OFF_Q  + 16777216;      // K pair-layout
static constexpr size_t OFF_V   = OFF_K  + 16777216;      // V pair-layout
static constexpr size_t OFF_SC  = OFF_V  + 16777216;      // (B*NH,S,S)
static constexpr size_t OFF_HS  = OFF_SC + 33554432;      // lstm h ping-pong 2dir*2*16384
static constexpr size_t OFF_BAR = OFF_HS + 65536;         // barrier words (16 uints, pad 64)
// K-pair-major transposed weights:
static constexpr size_t OFF_W0F = OFF_BAR + 64;           // 768*2048
static constexpr size_t OFF_W0B = OFF_W0F + 1572864;
static constexpr size_t OFF_W1F = OFF_W0B + 1572864;      // 1536*2048
static constexpr size_t OFF_W1B = OFF_W1F + 3145728;
static constexpr size_t OFF_IPT = OFF_W1B + 3145728;      // 1024*3072
static constexpr size_t OFF_OPT = OFF_IPT + 3145728;      // 1024*1024
static constexpr size_t OFF_H2T = OFF_OPT + 1048576;      // 1024*32
static constexpr size_t OFF_CWT = OFF_H2T + 32768;        // 1536*256
// reuse: CTX <- Y1, Z <- Q, EM <- K

// ---------------- global software barrier (per LSTM direction) ----------------
static __device__ __forceinline__ void dir_barrier(unsigned* cnt, unsigned* gen, unsigned nb) {
  __syncthreads();
  if (threadIdx.x == 0) {
    __threadfence();
    unsigned g = atomicAdd(gen, 0u);
    unsigned arrived = atomicAdd(cnt, 1u);
    if (arrived == nb - 1u) {
      atomicExch(cnt, 0u);
      __threadfence();
      atomicAdd(gen, 1u);
    } else {
      while (atomicAdd(gen, 0u) == g) { __builtin_amdgcn_s_sleep(1); }
    }
    __threadfence();
  }
  __syncthreads();
}

// ---------------- weight pre-transposes to K-pair-major ----------------
// out[(k>>1)*2N + n*2 + (k&1)] = W(n,k) : B-operand fetch becomes one coalesced b64.
__global__ void k_trw(const float* __restrict__ w, int N, int K, float* __restrict__ out) {
  int tid = blockIdx.x * blockDim.x + threadIdx.x;  // N*K threads
  int k = tid / N, n = tid - k * N;
  out[(size_t)(k >> 1) * 2 * N + n * 2 + (k & 1)] = w[(size_t)n * K + k];
}
// LSTM: concat [wih | whh] rows (4H x (in_dim+512)) -> pair-major, 2N = 4096.
__global__ void k_trlstm(const float* __restrict__ wih, const float* __restrict__ whh,
                         int in_dim, float* __restrict__ out) {
  int tid = blockIdx.x * blockDim.x + threadIdx.x;  // (in_dim+512)*2048 threads
  int k = tid >> 11, n = tid & 2047;
  float v = (k < in_dim) ? wih[(size_t)n * in_dim + k] : whh[(size_t)n * 512 + (k - in_dim)];
  out[(size_t)(k >> 1) * 4096 + n * 2 + (k & 1)] = v;
}
// conv_w (256,512,3) -> pair-major over k=(tap*512+cin), 2N = 512.
__global__ void k_trconv(const float* __restrict__ cw, float* __restrict__ out) {
  int tid = blockIdx.x * blockDim.x + threadIdx.x;  // 1536*256 threads
  int k = tid >> 8, n = tid & 255;
  int tap = k >> 9, cin = k & 511;
  out[(size_t)(k >> 1) * 512 + n * 2 + (k & 1)] = cw[((size_t)n * 512 + cin) * 3 + tap];
}

// ---------------- 1) embedding + feature proj + BN + ReLU + concat ----------------
__global__ void k_embed_feat(const int* __restrict__ ci, const float* __restrict__ feat,
                             const float* __restrict__ emb, const float* __restrict__ fpw,
                             const float* __restrict__ fpb, const float* __restrict__ bng,
                             const float* __restrict__ bnb, float* __restrict__ x) {
  int tid = blockIdx.x * blockDim.x + threadIdx.x;
  int bs = tid >> 8, e = tid & 255;
  int c = ci[bs];
  x[(size_t)bs * 512 + e] = emb[c * 256 + e];
  const float* f = feat + (size_t)bs * 64;
  const float* w = fpw + (size_t)e * 64;
  float acc = fpb[e];
#pragma unroll 8
  for (int k = 0; k < 64; k++) acc += f[k] * w[k];
  const float inv = 0.999995000037499688f;  // 1/sqrt(1+1e-5)
  acc = bng[e] * acc * inv + bnb[e];
  x[(size_t)bs * 512 + 256 + e] = fmaxf(acc, 0.0f);
}

// ---------------- 2) conv1d(k=3,pad=1) as implicit GEMM (branchless taps) ----------------
__global__ void k_conv(const float* __restrict__ x, const float* __restrict__ cwT,
                       const float* __restrict__ cb, float* __restrict__ h) {
  int wg = blockIdx.x * (blockDim.x >> 5) + (threadIdx.x >> 5);
  int lane = threadIdx.x & 31;
  int mt = wg >> 4, nt = wg & 15;
  int m0 = mt * 16, n0 = nt * 16;
  int mA = m0 + (lane & 15);
  int b = mA >> 9, s = mA & 511;
  int kl = (lane >> 4) * 2;
  int nB2 = (n0 + (lane & 15)) * 2;
  v8f acc = {0.f, 0.f, 0.f, 0.f, 0.f, 0.f, 0.f, 0.f};
#pragma unroll
  for (int tap = 0; tap < 3; ++tap) {
    int sp = s + tap - 1;
    bool ok = (unsigned)sp < 512u;
    const float* xrow = x + ((size_t)b * 512 + (ok ? sp : s)) * 512;
    float msk = ok ? 1.0f : 0.0f;
    const float* wt = cwT + (size_t)((tap * 512) >> 1) * 512;
    for (int c0 = 0; c0 < 512; c0 += 4) {
      int kk = c0 + kl;
      v2f a = *(const v2f*)(xrow + kk);
      a[0] *= msk; a[1] *= msk;
      v2f bb = *(const v2f*)(wt + (size_t)(kk >> 1) * 512 + nB2);
      acc = wmma4(a, bb, acc);
    }
  }
  float bias = cb[nB2 >> 1];
#pragma unroll
  for (int r = 0; r < 8; r++) {
    int m = m0 + r + ((lane >> 4) << 3);
    int br = m >> 9, sr = m & 511;
    h[((size_t)sr * 32 + br) * 256 + (nB2 >> 1)] = fmaxf(acc[r] + bias, 0.0f);
  }
}

// ---------------- 3) persistent BiLSTM layer ----------------
// 16 blocks x 256 thr: blocks 0-7 fwd, 8-15 bwd. 64 waves/dir, one (m,gate-n) tile pair each.
// h staged global->LDS via global_load_async_to_lds_b128 with XOR swizzle (bank-conflict-free
// A reads); cell state c lives in registers for all 512 steps.
__global__ void __launch_bounds__(256) k_lstm(
    const float* __restrict__ xin, float* __restrict__ yout,
    const float* __restrict__ wtf, const float* __restrict__ wtb,   // pair-major [K][4096]
    const float* __restrict__ bih0, const float* __restrict__ bhh0,
    const float* __restrict__ bih1, const float* __restrict__ bhh1,
    int in_dim, float* __restrict__ hstate, unsigned* __restrict__ bar) {
  __shared__ float sh[16384];  // 64KB swizzled h
  int dir = blockIdx.x >> 3;
  int blk = blockIdx.x & 7;
  int wave = (blk << 3) + (threadIdx.x >> 5);
  int lane = threadIdx.x & 31;
  const float* wt = dir ? wtb : wtf;
  const float* bih = dir ? bih1 : bih0;
  const float* bhh = dir ? bhh1 : bhh0;
  float* hb0 = hstate + (size_t)dir * 2 * 16384;
  float* hb1 = hb0 + 16384;
  unsigned* cnt = bar + dir * 2;
  unsigned* gen = bar + dir * 2 + 1;

  for (int i = threadIdx.x + blk * 256; i < 16384; i += 2048) { hb0[i] = 0.f; hb1[i] = 0.f; }
  dir_barrier(cnt, gen, 8);

  int mt = wave & 1, nt = wave >> 1;
  int m0 = mt * 16, n0 = nt * 16;
  int mA = m0 + (lane & 15);
  int kl = (lane >> 4) * 2;
  int nB = n0 + (lane & 15);
  int nB2 = nB * 2;
  unsigned swz = (unsigned)((mA & 15) << 2);
  float bI = bih[nB] + bhh[nB];
  float bF = bih[512 + nB] + bhh[512 + nB];
  float bG = bih[1024 + nB] + bhh[1024 + nB];
  float bO = bih[1536 + nB] + bhh[1536 + nB];
  float cst[8];
#pragma unroll
  for (int r = 0; r < 8; r++) cst[r] = 0.f;
  int dirOff = dir ? 512 : 0;
  unsigned ldsbase = (unsigned)(uintptr_t)(&sh[0]);

  for (int t = 0; t < 512; t++) {
    int tt = dir ? (511 - t) : t;
    const float* rb = (t & 1) ? hb1 : hb0;
    float* wb = (t & 1) ? hb0 : hb1;

    // --- async-stage h (64KB) into LDS with XOR swizzle; 16B chunks, 4096 total ---
#pragma unroll
    for (int i = 0; i < 16; ++i) {
      int c = threadIdx.x + (i << 8);
      int f = c << 2;                 // float offset in (32 x 512) h
      int m = f >> 9;
      int k = f & 511;
      unsigned dst = ldsbase + (unsigned)((m * 512 + (k ^ ((m & 15) << 2))) << 2);
      unsigned long long src = (unsigned long long)(uintptr_t)(rb + f);
      asm volatile("global_load_async_to_lds_b128 %0, %1, off"
                   :: "v"(dst), "v"(src) : "memory");
    }
    asm volatile("s_wait_asynccnt 0x0" ::: "memory");
    __syncthreads();

    v8f a0 = {0.f,0.f,0.f,0.f,0.f,0.f,0.f,0.f};
    v8f a1 = a0, a2 = a0, a3 = a0;
    // input contribution
    const float* xr = xin + ((size_t)tt * 32 + mA) * in_dim;
    for (int k0 = 0; k0 < in_dim; k0 += 4) {
      int kk = k0 + kl;
      v2f a = *(const v2f*)(xr + kk);
      const float* wk = wt + (size_t)(kk >> 1) * 4096 + nB2;
      a0 = wmma4(a, *(const v2f*)(wk), a0);
      a1 = wmma4(a, *(const v2f*)(wk + 1024), a1);
      a2 = wmma4(a, *(const v2f*)(wk + 2048), a2);
      a3 = wmma4(a, *(const v2f*)(wk + 3072), a3);
    }
    // recurrent contribution (A from swizzled LDS)
    for (int k0 = 0; k0 < 512; k0 += 4) {
      int kk = k0 + kl;
      v2f a = *(const v2f*)(&sh[(size_t)mA * 512 + (kk ^ swz)]);
      const float* wk = wt + (size_t)((in_dim + kk) >> 1) * 4096 + nB2;
      a0 = wmma4(a, *(const v2f*)(wk), a0);
      a1 = wmma4(a, *(const v2f*)(wk + 1024), a1);
      a2 = wmma4(a, *(const v2f*)(wk + 2048), a2);
      a3 = wmma4(a, *(const v2f*)(wk + 3072), a3);
    }
#pragma unroll
    for (int r = 0; r < 8; r++) {
      float ig = sig_(a0[r] + bI);
      float fg = sig_(a1[r] + bF);
      float gg = tanhf(a2[r] + bG);
      float og = sig_(a3[r] + bO);
      float c = fg * cst[r] + ig * gg;
      cst[r] = c;
      float hv = og * tanhf(c);
      int m = m0 + r + ((lane >> 4) << 3);
      wb[(size_t)m * 512 + nB] = hv;
      yout[((size_t)tt * 32 + m) * 1024 + dirOff + nB] = hv;
    }
    dir_barrier(cnt, gen, 8);  // publish h for next step
  }
}

// ---------------- 4) QKV projection ----------------
__global__ void k_qkv(const float* __restrict__ y2, const float* __restrict__ wT,
                      const float* __restrict__ bias, float* __restrict__ q,
                      float* __restrict__ kp, float* __restrict__ vp) {
  int wg = blockIdx.x * (blockDim.x >> 5) + (threadIdx.x >> 5);
  int lane = threadIdx.x & 31;
  int mt = wg / 192, nt = wg % 192;
  int m0 = mt * 16, n0 = nt * 16;
  int mA = m0 + (lane & 15);
  int kl = (lane >> 4) * 2;
  int nB = n0 + (lane & 15);
  const float* ar = y2 + (size_t)mA * 1024;
  v8f acc = {0.f,0.f,0.f,0.f,0.f,0.f,0.f,0.f};
  for (int k0 = 0; k0 < 1024; k0 += 4) {
    int kk = k0 + kl;
    acc = wmma4(*(const v2f*)(ar + kk), *(const v2f*)(wT + (size_t)(kk >> 1) * 6144 + nB * 2), acc);
  }
  float bs = bias[nB];
  int which = nB >> 10;
  int cc = nB & 1023;
  int hh = cc >> 8, dd = cc & 255;
#pragma unroll
  for (int r = 0; r < 8; r++) {
    int m = m0 + r + ((lane >> 4) << 3);
    int s = m >> 5, b = m & 31;
    size_t bh = (size_t)(b * 4 + hh) * 131072;
    float val = acc[r] + bs;
    if (which == 0)      q[((size_t)(b * 4 + hh) * 512 + s) * 256 + dd] = val;
    else if (which == 1) kp[bh + (size_t)(dd >> 1) * 1024 + s * 2 + (dd & 1)] = val;   // pair over d
    else                 vp[bh + (size_t)(s >> 1) * 512 + dd * 2 + (s & 1)] = val;     // pair over s
  }
}

// ---------------- 5) scores = q k^T / 16, masked ----------------
__global__ void k_scores(const float* __restrict__ q, const float* __restrict__ kp,
                         const unsigned char* __restrict__ mask, float* __restrict__ sc) {
  int wg = blockIdx.x * (blockDim.x >> 5) + (threadIdx.x >> 5);
  int lane = threadIdx.x & 31;
  int bh = wg >> 10;
  int tid = wg & 1023;
  int mt = tid >> 5, nt = tid & 31;
  int m0 = mt * 16, n0 = nt * 16;
  int b = bh >> 2;
  int mA = m0 + (lane & 15);
  int kl = (lane >> 4) * 2;
  int nB = n0 + (lane & 15);
  const float* ar = q + (size_t)bh * 131072 + (size_t)mA * 256;
  const float* wr = kp + (size_t)bh * 131072;
  v8f acc = {0.f,0.f,0.f,0.f,0.f,0.f,0.f,0.f};
  for (int k0 = 0; k0 < 256; k0 += 4) {
    int kk = k0 + kl;
    acc = wmma4(*(const v2f*)(ar + kk), *(const v2f*)(wr + (size_t)(kk >> 1) * 1024 + nB * 2), acc);
  }
  int mv = mask[b * 512 + nB];
#pragma unroll
  for (int r = 0; r < 8; r++) {
    int m = m0 + r + ((lane >> 4) << 3);
    sc[((size_t)bh * 512 + m) * 512 + nB] = mv ? acc[r] * 0.0625f : -1.0e9f;
  }
}

// ---------------- 6) row softmax over 512 keys ----------------
__global__ void k_softmax(float* __restrict__ sc) {
  __shared__ float red[128];
  float* row = sc + (size_t)blockIdx.x * 512;
  int t = threadIdx.x;
  float mx = -1e30f;
  for (int i = t; i < 512; i += 128) mx = fmaxf(mx, row[i]);
  red[t] = mx; __syncthreads();
  for (int s = 64; s > 0; s >>= 1) { if (t < s) red[t] = fmaxf(red[t], red[t + s]); __syncthreads(); }
  mx = red[0]; __syncthreads();
  float sum = 0.f;
  for (int i = t; i < 512; i += 128) { float e = expf(row[i] - mx); row[i] = e; sum += e; }
  red[t] = sum; __syncthreads();
  for (int s = 64; s > 0; s >>= 1) { if (t < s) red[t] += red[t + s]; __syncthreads(); }
  float inv = 1.0f / red[0];
  for (int i = t; i < 512; i += 128) row[i] *= inv;
}

// ---------------- 7) ctx = probs @ V ----------------
__global__ void k_ctx(const float* __restrict__ p, const float* __restrict__ vp,
                      float* __restrict__ ctx) {
  int wg = blockIdx.x * (blockDim.x >> 5) + (threadIdx.x >> 5);
  int lane = threadIdx.x & 31;
  int bh = wg >> 9;
  int tid = wg & 511;
  int mt = tid >> 4, nt = tid & 15;
  int m0 = mt * 16, n0 = nt * 16;
  int mA = m0 + (lane & 15);
  int kl = (lane >> 4) * 2;
  int nB = n0 + (lane & 15);
  const float* ar = p + (size_t)bh * 262144 + (size_t)mA * 512;
  const float* vr = vp + (size_t)bh * 131072;
  v8f acc = {0.f,0.f,0.f,0.f,0.f,0.f,0.f,0.f};
  for (int k0 = 0; k0 < 512; k0 += 4) {
    int kk = k0 + kl;
    acc = wmma4(*(const v2f*)(ar + kk), *(const v2f*)(vr + (size_t)(kk >> 1) * 512 + nB * 2), acc);
  }
  int b = bh >> 2, hh = bh & 3;
#pragma unroll
  for (int r = 0; r < 8; r++) {
    int s = m0 + r + ((lane >> 4) << 3);
    ctx[((size_t)s * 32 + b) * 1024 + hh * 256 + nB] = acc[r];
  }
}

// ---------------- 8) out proj + residual ----------------
__global__ void k_outproj(const float* __restrict__ ctx, const float* __restrict__ wT,
                          const float* __restrict__ bias, const float* __restrict__ y2,
                          float* __restrict__ z) {
  int wg = blockIdx.x * (blockDim.x >> 5) + (threadIdx.x >> 5);
  int lane = threadIdx.x & 31;
  int mt = wg >> 6, nt = wg & 63;
  int m0 = mt * 16, n0 = nt * 16;
  int mA = m0 + (lane & 15);
  int kl = (lane >> 4) * 2;
  int nB = n0 + (lane & 15);
  const float* ar = ctx + (size_t)mA * 1024;
  v8f acc = {0.f,0.f,0.f,0.f,0.f,0.f,0.f,0.f};
  for (int k0 = 0; k0 < 1024; k0 += 4) {
    int kk = k0 + kl;
    acc = wmma4(*(const v2f*)(ar + kk), *(const v2f*)(wT + (size_t)(kk >> 1) * 2048 + nB * 2), acc);
  }
  float bs = bias[nB];
#pragma unroll
  for (int r = 0; r < 8; r++) {
    int m = m0 + r + ((lane >> 4) << 3);
    z[(size_t)m * 1024 + nB] = acc[r] + bs + y2[(size_t)m * 1024 + nB];
  }
}

// ---------------- 9) LayerNorm (in place) ----------------
__global__ void k_ln(float* __restrict__ z, const float* __restrict__ g,
                     const float* __restrict__ bta) {
  __shared__ float r1[256], r2[256];
  float* row = z + (size_t)blockIdx.x * 1024;
  int t = threadIdx.x;
  float vals[4];
  float s = 0.f, s2 = 0.f;
#pragma unroll
  for (int i = 0; i < 4; i++) { float v = row[t + 256 * i]; vals[i] = v; s += v; s2 += v * v; }
  r1[t] = s; r2[t] = s2; __syncthreads();
  for (int st = 128; st > 0; st >>= 1) {
    if (t < st) { r1[t] += r1[t + st]; r2[t] += r2[t + st]; }
    __syncthreads();
  }
  float mu = r1[0] * (1.0f / 1024.0f);
  float var = r2[0] * (1.0f / 1024.0f) - mu * mu;
  float inv = rsqrtf(var + 1e-5f);
#pragma unroll
  for (int i = 0; i < 4; i++) {
    int d = t + 256 * i;
    row[d] = (vals[i] - mu) * inv * g[d] + bta[d];
  }
}

// ---------------- 10) emissions ----------------
__global__ void k_emis(const float* __restrict__ z, const float* __restrict__ wT,
                       const float* __restrict__ bias, float* __restrict__ em) {
  int wg = blockIdx.x * (blockDim.x >> 5) + (threadIdx.x >> 5);
  int lane = threadIdx.x & 31;
  int mt = wg >> 1, nt = wg & 1;
  int m0 = mt * 16, n0 = nt * 16;
  int mA = m0 + (lane & 15);
  int kl = (lane >> 4) * 2;
  int nB = n0 + (lane & 15);
  const float* ar = z + (size_t)mA * 1024;
  v8f acc = {0.f,0.f,0.f,0.f,0.f,0.f,0.f,0.f};
  for (int k0 = 0; k0 < 1024; k0 += 4) {
    int kk = k0 + kl;
    acc = wmma4(*(const v2f*)(ar + kk), *(const v2f*)(wT + (size_t)(kk >> 1) * 64 + nB * 2), acc);
  }
  float bs = bias[nB];
#pragma unroll
  for (int r = 0; r < 8; r++) {
    int m = m0 + r + ((lane >> 4) << 3);
    int sdx = m >> 5, b = m & 31;
    em[((size_t)b * 512 + sdx) * 32 + nB] = acc[r] + bs;
  }
}

// ---------------- 11) CRF ----------------
__global__ void __launch_bounds__(1024) k_crf(
    const float* __restrict__ em, const int* __restrict__ tags,
    const unsigned char* __restrict__ mask, const float* __restrict__ start,
    const float* __restrict__ endv, const float* __restrict__ trans,
    float* __restrict__ out) {
  __shared__ float sc[32][32];
  __shared__ float num[32];
  __shared__ float llh[32];
  int b = threadIdx.x >> 5, j = threadIdx.x & 31;
  sc[b][j] = start[j] + em[(size_t)b * 512 * 32 + j];
  if (j == 0) {
    int t0 = tags[b * 512];
    float na = start[t0] + em[(size_t)b * 512 * 32 + t0];
    int prev = t0;
    int cntm = mask[b * 512] ? 1 : 0;
    for (int t = 1; t < 512; t++) {
      int tg = tags[b * 512 + t];
      if (mask[b * 512 + t]) {
        na += trans[prev * 32 + tg] + em[((size_t)b * 512 + t) * 32 + tg];
        cntm++;
      }
      prev = tg;
    }
    na += endv[tags[b * 512 + (cntm - 1)]];
    num[b] = na;
  }
  __syncthreads();
  for (int t = 1; t < 512; t++) {
    float mx = -1e30f;
#pragma unroll
    for (int i = 0; i < 32; i++) mx = fmaxf(mx, sc[b][i] + trans[i * 32 + j]);
    float s = 0.f;
#pragma unroll
    for (int i = 0; i < 32; i++) s += expf(sc[b][i] + trans[i * 32 + j] - mx);
    float nxt = mx + logf(s) + em[((size_t)b * 512 + t) * 32 + j];
    float nv = mask[b * 512 + t] ? nxt : sc[b][j];
    __syncthreads();
    sc[b][j] = nv;
    __syncthreads();
  }
  sc[b][j] += endv[j];
  __syncthreads();
  if (j == 0) {
    float mx = -1e30f;
    for (int i = 0; i < 32; i++) mx = fmaxf(mx, sc[b][i]);
    float s = 0.f;
    for (int i = 0; i < 32; i++) s += expf(sc[b][i] - mx);
    llh[b] = num[b] - (mx + logf(s));
  }
  __syncthreads();
  if (threadIdx.x == 0) {
    float s = 0.f;
    for (int i = 0; i < 32; i++) s += llh[i];
    out[0] = -s * (1.0f / 32.0f);
  }
}

// ---------------- host launcher ----------------
extern "C" void kernel_launch(void* const* d_in, const int* in_sizes, int n_in,
                              void* d_out, int out_size, void* d_ws, size_t ws_size,
                              hipStream_t stream) {
  (void)in_sizes; (void)n_in; (void)out_size; (void)ws_size;
  const int* ci = (const int*)d_in[0];
  const float* feat = (const float*)d_in[1];
  const unsigned char* mask = (const unsigned char*)d_in[2];
  const int* tags = (const int*)d_in[3];
  const float* emb = (const float*)d_in[4];
  const float* fpw = (const float*)d_in[5];
  const float* fpb = (const float*)d_in[6];
  const float* bng = (const float*)d_in[7];
  const float* bnb = (const float*)d_in[8];
  const float* cw = (const float*)d_in[9];
  const float* cb = (const float*)d_in[10];
  const float* l0f_wih = (const float*)d_in[11];
  const float* l0f_whh = (const float*)d_in[12];
  const float* l0f_bih = (const float*)d_in[13];
  const float* l0f_bhh = (const float*)d_in[14];
  const float* l0b_wih = (const float*)d_in[15];
  const float* l0b_whh = (const float*)d_in[16];
  const float* l0b_bih = (const float*)d_in[17];
  const float* l0b_bhh = (const float*)d_in[18];
  const float* l1f_wih = (const float*)d_in[19];
  const float* l1f_whh = (const float*)d_in[20];
  const float* l1f_bih = (const float*)d_in[21];
  const float* l1f_bhh = (const float*)d_in[22];
  const float* l1b_wih = (const float*)d_in[23];
  const float* l1b_whh = (const float*)d_in[24];
  const float* l1b_bih = (const float*)d_in[25];
  const float* l1b_bhh = (const float*)d_in[26];
  const float* ipw = (const float*)d_in[27];
  const float* ipb = (const float*)d_in[28];
  const float* opw = (const float*)d_in[29];
  const float* opb = (const float*)d_in[30];
  const float* lng = (const float*)d_in[31];
  const float* lnb = (const float*)d_in[32];
  const float* h2tw = (const float*)d_in[33];
  const float* h2tb = (const float*)d_in[34];
  const float* crs = (const float*)d_in[35];
  const float* cre = (const float*)d_in[36];
  const float* crt = (const float*)d_in[37];
  float* out = (float*)d_out;
  float* ws = (float*)d_ws;

  float* X = ws + OFF_X;
  float* HB = ws + OFF_HB;
  float* Y1 = ws + OFF_Y1;
  float* Y2 = ws + OFF_Y2;
  float* Q = ws + OFF_Q;
  float* Kp = ws + OFF_K;
  float* Vp = ws + OFF_V;
  float* SC = ws + OFF_SC;
  float* HS = ws + OFF_HS;
  unsigned* BAR = (unsigned*)(ws + OFF_BAR);
  float* W0F = ws + OFF_W0F;
  float* W0B = ws + OFF_W0B;
  float* W1F = ws + OFF_W1F;
  float* W1B = ws + OFF_W1B;
  float* IPT = ws + OFF_IPT;
  float* OPT = ws + OFF_OPT;
  float* H2T = ws + OFF_H2T;
  float* CWT = ws + OFF_CWT;
  float* CTX = Y1;  // reuse
  float* Z = Q;     // reuse
  float* EM = Kp;   // reuse

  hipMemsetAsync(BAR, 0, 64, stream);

  // one-time (per call) weight reshapes
  k_trconv<<<1536, 256, 0, stream>>>(cw, CWT);
  k_trlstm<<<6144, 256, 0, stream>>>(l0f_wih, l0f_whh, 256, W0F);
  k_trlstm<<<6144, 256, 0, stream>>>(l0b_wih, l0b_whh, 256, W0B);
  k_trlstm<<<12288, 256, 0, stream>>>(l1f_wih, l1f_whh, 1024, W1F);
  k_trlstm<<<12288, 256, 0, stream>>>(l1b_wih, l1b_whh, 1024, W1B);
  k_trw<<<12288, 256, 0, stream>>>(ipw, 3072, 1024, IPT);
  k_trw<<<4096, 256, 0, stream>>>(opw, 1024, 1024, OPT);
  k_trw<<<128, 256, 0, stream>>>(h2tw, 32, 1024, H2T);

  k_embed_feat<<<16384, 256, 0, stream>>>(ci, feat, emb, fpw, fpb, bng, bnb, X);
  k_conv<<<2048, 256, 0, stream>>>(X, CWT, cb, HB);
  k_lstm<<<16, 256, 0, stream>>>(HB, Y1, W0F, W0B, l0f_bih, l0f_bhh, l0b_bih, l0b_bhh,
                                 256, HS, BAR);
  k_lstm<<<16, 256, 0, stream>>>(Y1, Y2, W1F, W1B, l1f_bih, l1f_bhh, l1b_bih, l1b_bhh,
                                 1024, HS, BAR);
  k_qkv<<<24576, 256, 0, stream>>>(Y2, IPT, ipb, Q, Kp, Vp);
  k_scores<<<16384, 256, 0, stream>>>(Q, Kp, mask, SC);
  k_softmax<<<65536, 128, 0, stream>>>(SC);
  k_ctx<<<8192, 256, 0, stream>>>(SC, Vp, CTX);
  k_outproj<<<8192, 256, 0, stream>>>(CTX, OPT, opb, Y2, Z);
  k_ln<<<16384, 256, 0, stream>>>(Z, lng, lnb);
  k_emis<<<256, 256, 0, stream>>>(Z, H2T, h2tb, EM);
  k_crf<<<1, 1024, 0, stream>>>(EM, tags, mask, crs, cre, crt, out);
}